// GIN5_18537078849979
// MI455X (gfx1250) — compile-verified
//
#include <hip/hip_runtime.h>

#define N_NODES 100000
#define N_EDGES 600000
#define DFEAT 128
#define LAYERS 5
#define BN_EPS 1e-5f
#define ZSTR 132   // padded LDS row stride (floats) to avoid bank conflicts

typedef __attribute__((ext_vector_type(2))) float v2f;
typedef __attribute__((ext_vector_type(8))) float v8f;

// ---------------- kernel 1: agg = (1+eps) * h  (GIN self term) ----------------
__global__ void GIN_init_agg(const float4* __restrict__ h4,
                             float4* __restrict__ agg,
                             const float* __restrict__ epsP, int n4) {
    int i = blockIdx.x * blockDim.x + threadIdx.x;
    if (i >= n4) return;
    float e = 1.0f + epsP[0];
    float4 v = h4[i];
    agg[i] = make_float4(e * v.x, e * v.y, e * v.z, e * v.w);
}

// ---------------- kernel 2: edge scatter-add (one wave32 per edge) ----------------
__global__ void GIN_scatter(const float* __restrict__ h,
                            const long long* __restrict__ src,
                            const long long* __restrict__ dst,
                            float* __restrict__ agg) {
    int t = blockIdx.x * blockDim.x + threadIdx.x;
    int e    = t >> 5;
    int lane = t & 31;
    if (e >= N_EDGES) return;
    int s = (int)src[e];
    int d = (int)dst[e];
    float4 hv = *(const float4*)(h + (size_t)s * DFEAT + lane * 4);
    float* ap = agg + (size_t)d * DFEAT + lane * 4;
    atomicAdd(ap + 0, hv.x);
    atomicAdd(ap + 1, hv.y);
    atomicAdd(ap + 2, hv.z);
    atomicAdd(ap + 3, hv.w);
}

// ---------------- TDM: DMA a 128x128 fp32 tile (row-major, contiguous) into LDS ----
__device__ __forceinline__ void tdm_load_w(const float* __restrict__ Wsrc,
                                           float* __restrict__ wdst) {
#if __has_builtin(__builtin_amdgcn_tensor_load_to_lds)
    typedef __attribute__((ext_vector_type(4))) unsigned int u32x4;
    typedef __attribute__((ext_vector_type(8))) int i32x8;
    typedef __attribute__((ext_vector_type(4))) int i32x4;
    unsigned lds_off = (unsigned)(unsigned long long)wdst;   // low 32 bits = LDS offset
    unsigned long long ga = (unsigned long long)Wsrc;
    // D# group 0: count=1 | lds_addr | global_addr[56:0] | type=2
    u32x4 g0 = { 1u,
                 lds_off,
                 (unsigned)(ga & 0xFFFFFFFFull),
                 (unsigned)((ga >> 32) & 0x01FFFFFFull) | (2u << 30) };
    // D# group 1: data_size=4B; tensor 128x128; tile 128x128; stride0=128; stride1=16384
    i32x8 g1 = { (int)(2u << 16),            // workgroup_mask=0, data_size=2 (4B)
                 (int)(128u << 16),          // tensor_dim0 low16 at bits 63:48
                 (int)(128u << 16),          // tensor_dim0 hi=0, tensor_dim1 low16 at 95:80
                 (int)(128u << 16),          // tensor_dim1 hi=0, tile_dim0 at 127:112
                 (int)128,                   // tile_dim1=128, tile_dim2=0
                 (int)128,                   // tensor_dim0_stride low32
                 (int)(16384u << 16),        // stride0 hi=0, tensor_dim1_stride low16
                 0 };
    i32x4 gz4 = { 0, 0, 0, 0 };
    i32x8 gz8 = { 0, 0, 0, 0, 0, 0, 0, 0 };
    // 6-arg form (clang-23 / amdgpu-toolchain)
    __builtin_amdgcn_tensor_load_to_lds(g0, g1, gz4, gz4, gz8, 0);
#else
    (void)Wsrc; (void)wdst;
#endif
}

__device__ __forceinline__ void tdm_wait() {
#if __has_builtin(__builtin_amdgcn_s_wait_tensorcnt)
    __builtin_amdgcn_s_wait_tensorcnt(0);
#endif
}

// ---------------- kernel 3: fused GIN MLP via WMMA fp32 ----------------
// block = 256 threads = 8 waves; each wave owns a 16-row tile; block covers 128 rows.
__global__ __launch_bounds__(256, 1)
void GIN_mlp(const float* __restrict__ agg,   // z = (1+eps)*h + sum already accumulated
             const float* __restrict__ W1, const float* __restrict__ b1,
             const float* __restrict__ g1, const float* __restrict__ bb1,
             const float* __restrict__ m1, const float* __restrict__ va1,
             const float* __restrict__ W2, const float* __restrict__ b2,
             const float* __restrict__ g2, const float* __restrict__ bb2,
             const float* __restrict__ m2, const float* __restrict__ va2,
             float* __restrict__ out) {
    extern __shared__ float smem[];
    float* wlds = smem;                 // DFEAT*DFEAT floats (64 KB), W1 then W2
    float* zlds = smem + DFEAT * DFEAT; // 128 rows * ZSTR floats (66 KB)

    const int tid   = threadIdx.x;
    const int lane  = tid & 31;
    const int wave  = tid >> 5;
    const int hsel  = lane >> 4;   // 0: lanes 0-15, 1: lanes 16-31
    const int nl    = lane & 15;
    const int mrow  = wave * 16;   // local row base for this wave's tile
    const int rowBase = blockIdx.x * 128;

#if __has_builtin(__builtin_amdgcn_tensor_load_to_lds)
    // async DMA W1 -> LDS via Tensor Data Mover (overlaps the z staging below)
    if (wave == 0) tdm_load_w(W1, wlds);
#else
    {
        const float4* wsrc = (const float4*)W1;
        float4* wdst = (float4*)wlds;
        #pragma unroll 4
        for (int i = tid; i < DFEAT * DFEAT / 4; i += 256) wdst[i] = wsrc[i];
    }
#endif
    // stage z tile into LDS (row-clamped at the tail block)
    {
        const float4* a4 = (const float4*)agg;
        #pragma unroll 4
        for (int i = tid; i < 128 * DFEAT / 4; i += 256) {
            int r  = i >> 5;         // DFEAT/4 == 32 float4 per row
            int c4 = i & 31;
            int gr = rowBase + r; if (gr >= N_NODES) gr = N_NODES - 1;
            float4 z = a4[(size_t)gr * 32 + c4];
            *(float4*)(zlds + r * ZSTR + c4 * 4) = z;
        }
    }
#if __has_builtin(__builtin_amdgcn_tensor_load_to_lds)
    if (wave == 0) tdm_wait();
#endif
    __syncthreads();

    v8f acc[8];
    const v8f vzero = {0.f, 0.f, 0.f, 0.f, 0.f, 0.f, 0.f, 0.f};

    // ---------------- GEMM 1: z @ W1 ----------------
    #pragma unroll
    for (int n = 0; n < 8; n++) acc[n] = vzero;

    for (int k0 = 0; k0 < DFEAT; k0 += 4) {
        int kk = k0 + 2 * hsel;
        // A fragment: 16x4 fp32, lane (hsel,nl) holds row nl, K = kk, kk+1
        v2f a = *(const v2f*)(zlds + (mrow + nl) * ZSTR + kk);
        #pragma unroll
        for (int n = 0; n < 8; n++) {
            v2f b;
            b.x = wlds[kk * DFEAT + n * 16 + nl];
            b.y = wlds[(kk + 1) * DFEAT + n * 16 + nl];
            acc[n] = __builtin_amdgcn_wmma_f32_16x16x4_f32(
                false, a, false, b, (short)0, acc[n], false, false);
        }
    }

    // epilogue 1: +b1, ReLU, BN1 -> write intermediate back to own zlds region
    #pragma unroll
    for (int n = 0; n < 8; n++) {
        int col = n * 16 + nl;
        float sc   = g1[col] * rsqrtf(va1[col] + BN_EPS);
        float sh   = bb1[col] - m1[col] * sc;
        float bias = b1[col];
        #pragma unroll
        for (int j = 0; j < 8; j++) {
            int row = mrow + j + hsel * 8;
            float v = acc[n][j] + bias;
            v = fmaxf(v, 0.0f);
            v = v * sc + sh;
            zlds[row * ZSTR + col] = v;
        }
    }
    __syncthreads();

    // stage W2 (reuse the W LDS buffer)
#if __has_builtin(__builtin_amdgcn_tensor_load_to_lds)
    if (wave == 0) { tdm_load_w(W2, wlds); tdm_wait(); }
#else
    {
        const float4* wsrc = (const float4*)W2;
        float4* wdst = (float4*)wlds;
        #pragma unroll 4
        for (int i = tid; i < DFEAT * DFEAT / 4; i += 256) wdst[i] = wsrc[i];
    }
#endif
    __syncthreads();

    // ---------------- GEMM 2: y1 @ W2 ----------------
    #pragma unroll
    for (int n = 0; n < 8; n++) acc[n] = vzero;

    for (int k0 = 0; k0 < DFEAT; k0 += 4) {
        int kk = k0 + 2 * hsel;
        v2f a = *(const v2f*)(zlds + (mrow + nl) * ZSTR + kk);
        #pragma unroll
        for (int n = 0; n < 8; n++) {
            v2f b;
            b.x = wlds[kk * DFEAT + n * 16 + nl];
            b.y = wlds[(kk + 1) * DFEAT + n * 16 + nl];
            acc[n] = __builtin_amdgcn_wmma_f32_16x16x4_f32(
                false, a, false, b, (short)0, acc[n], false, false);
        }
    }

    // epilogue 2: +b2, BN2, ReLU -> global store (bounds-checked)
    #pragma unroll
    for (int n = 0; n < 8; n++) {
        int col = n * 16 + nl;
        float sc   = g2[col] * rsqrtf(va2[col] + BN_EPS);
        float sh   = bb2[col] - m2[col] * sc;
        float bias = b2[col];
        #pragma unroll
        for (int j = 0; j < 8; j++) {
            int row = mrow + j + hsel * 8;
            int gr  = rowBase + row;
            float v = acc[n][j] + bias;
            v = v * sc + sh;
            v = fmaxf(v, 0.0f);
            if (gr < N_NODES) out[(size_t)gr * DFEAT + col] = v;
        }
    }
}

extern "C" void kernel_launch(void* const* d_in, const int* in_sizes, int n_in,
                              void* d_out, int out_size, void* d_ws, size_t ws_size,
                              hipStream_t stream) {
    (void)in_sizes; (void)n_in; (void)out_size; (void)ws_size;

    const float*      x   = (const float*)d_in[0];
    const long long*  ei  = (const long long*)d_in[1];
    const float*      W1  = (const float*)d_in[2];
    const float*      b1  = (const float*)d_in[3];
    const float*      g1  = (const float*)d_in[4];
    const float*      bb1 = (const float*)d_in[5];
    const float*      m1  = (const float*)d_in[6];
    const float*      va1 = (const float*)d_in[7];
    const float*      W2  = (const float*)d_in[8];
    const float*      b2  = (const float*)d_in[9];
    const float*      eps = (const float*)d_in[10];
    const float*      g2  = (const float*)d_in[11];
    const float*      bb2 = (const float*)d_in[12];
    const float*      m2  = (const float*)d_in[13];
    const float*      va2 = (const float*)d_in[14];

    float* agg  = (float*)d_ws;
    float* hA   = agg + (size_t)N_NODES * DFEAT;
    float* outp = (float*)d_out;

    const long long* src = ei;
    const long long* dst = ei + N_EDGES;

    const int smemBytes = (DFEAT * DFEAT + 128 * ZSTR) * 4;
    const int n4 = N_NODES * DFEAT / 4;
    const long long scatterThreads = (long long)N_EDGES * 32;

    const float* hin = x;
    for (int i = 0; i < LAYERS; i++) {
        // ping-pong so the last (even) layer lands in d_out
        float* hout = (i & 1) ? hA : outp;

        GIN_init_agg<<<(n4 + 255) / 256, 256, 0, stream>>>(
            (const float4*)hin, (float4*)agg, eps + i, n4);

        GIN_scatter<<<(int)((scatterThreads + 255) / 256), 256, 0, stream>>>(
            hin, src, dst, agg);

        GIN_mlp<<<(N_NODES + 127) / 128, 256, smemBytes, stream>>>(
            agg,
            W1 + (size_t)i * DFEAT * DFEAT, b1 + i * DFEAT,
            g1 + i * DFEAT, bb1 + i * DFEAT, m1 + i * DFEAT, va1 + i * DFEAT,
            W2 + (size_t)i * DFEAT * DFEAT, b2 + i * DFEAT,
            g2 + i * DFEAT, bb2 + i * DFEAT, m2 + i * DFEAT, va2 + i * DFEAT,
            hout);

        hin = hout;
    }
}